// SigLoss_86457691668860
// MI455X (gfx1250) — compile-verified
//
#include <hip/hip_runtime.h>
#include <math.h>

// CDNA5 / gfx1250: wave32, WMMA 16x16x4 F32.
typedef __attribute__((ext_vector_type(2))) float v2f;
typedef __attribute__((ext_vector_type(8))) float v8f;

#ifndef SIG_D
#define SIG_D 1024            // embedding dim (fixed by reference)
#endif
#define NWAVES 8              // waves per 16-row tile (K-split)
#define KSPAN  (SIG_D / NWAVES)   // 128 k-values per wave

// Block = 256 threads = 8 waves, one 16-row tile per block.
// Each wave accumulates a K-slice of three WMMA products:
//   acc_it = IMG*TXT^T (diag -> img_i.txt_i)
//   acc_ii = IMG*IMG^T (diag -> ||img_i||^2)
//   acc_tt = TXT*TXT^T (diag -> ||txt_i||^2)
// Partial diagonals are combined across waves in LDS in a fixed order
// (deterministic), then 16 per-row losses are reduced to one partial/block.
__global__ __launch_bounds__(32 * NWAVES)
void siglip_diag_wmma(const float* __restrict__ img,
                      const float* __restrict__ txt,
                      const float* __restrict__ t_prime,
                      const float* __restrict__ bias,
                      float* __restrict__ partial)
{
    const int tid  = threadIdx.x;
    const int lane = tid & 31;
    const int wave = tid >> 5;
    const int r0   = blockIdx.x << 4;          // first row of tile
    const int m    = lane & 15;                // frag row
    const int koff = (lane >> 4) << 1;         // lanes 0-15 -> K{0,1}; 16-31 -> K{2,3}
    const int kbase = wave * KSPAN;            // this wave's K slice

    const float* ip = img + (size_t)(r0 + m) * SIG_D + kbase + koff;
    const float* tp = txt + (size_t)(r0 + m) * SIG_D + kbase + koff;

    v8f acc_it = {};
    v8f acc_ii = {};
    v8f acc_tt = {};

    // Batch 4 k-steps (16 k): 8 loads issued back-to-back, then 12 WMMAs,
    // keeping ~8 loads outstanding per wave for memory-level parallelism.
#pragma unroll
    for (int k = 0; k < KSPAN; k += 16) {
        v2f a0 = *(const v2f*)(ip + k);
        v2f b0 = *(const v2f*)(tp + k);
        v2f a1 = *(const v2f*)(ip + k + 4);
        v2f b1 = *(const v2f*)(tp + k + 4);
        v2f a2 = *(const v2f*)(ip + k + 8);
        v2f b2 = *(const v2f*)(tp + k + 8);
        v2f a3 = *(const v2f*)(ip + k + 12);
        v2f b3 = *(const v2f*)(tp + k + 12);

        acc_it = __builtin_amdgcn_wmma_f32_16x16x4_f32(false, a0, false, b0, (short)0, acc_it, false, false);
        acc_ii = __builtin_amdgcn_wmma_f32_16x16x4_f32(false, a0, false, a0, (short)0, acc_ii, false, false);
        acc_tt = __builtin_amdgcn_wmma_f32_16x16x4_f32(false, b0, false, b0, (short)0, acc_tt, false, false);

        acc_it = __builtin_amdgcn_wmma_f32_16x16x4_f32(false, a1, false, b1, (short)0, acc_it, false, false);
        acc_ii = __builtin_amdgcn_wmma_f32_16x16x4_f32(false, a1, false, a1, (short)0, acc_ii, false, false);
        acc_tt = __builtin_amdgcn_wmma_f32_16x16x4_f32(false, b1, false, b1, (short)0, acc_tt, false, false);

        acc_it = __builtin_amdgcn_wmma_f32_16x16x4_f32(false, a2, false, b2, (short)0, acc_it, false, false);
        acc_ii = __builtin_amdgcn_wmma_f32_16x16x4_f32(false, a2, false, a2, (short)0, acc_ii, false, false);
        acc_tt = __builtin_amdgcn_wmma_f32_16x16x4_f32(false, b2, false, b2, (short)0, acc_tt, false, false);

        acc_it = __builtin_amdgcn_wmma_f32_16x16x4_f32(false, a3, false, b3, (short)0, acc_it, false, false);
        acc_ii = __builtin_amdgcn_wmma_f32_16x16x4_f32(false, a3, false, a3, (short)0, acc_ii, false, false);
        acc_tt = __builtin_amdgcn_wmma_f32_16x16x4_f32(false, b3, false, b3, (short)0, acc_tt, false, false);
    }

    // Diagonal of 16x16 f32 C/D layout:
    //   lanes 0-15:  (M=v,   N=lane)    -> diag at v = lane      (lanes 0..7)
    //   lanes 16-31: (M=8+v, N=lane-16) -> diag at v = lane - 24 (lanes 24..31)
    int idx = -1;
    if (lane < 8)        idx = lane;
    else if (lane >= 24) idx = lane - 24;

    float it = 0.f, ii = 0.f, tt = 0.f;
#pragma unroll
    for (int j = 0; j < 8; ++j) {              // constant-index selects
        it = (j == idx) ? acc_it[j] : it;
        ii = (j == idx) ? acc_ii[j] : ii;
        tt = (j == idx) ? acc_tt[j] : tt;
    }

    // Per-wave partial diagonals -> LDS slots (one slot per wave: no races,
    // fixed-order combine below keeps the result deterministic).
    __shared__ float sIT[NWAVES][16];
    __shared__ float sII[NWAVES][16];
    __shared__ float sTT[NWAVES][16];
    if (idx >= 0) {
        const int row = (lane < 8) ? lane : (lane - 16);
        sIT[wave][row] = it;
        sII[wave][row] = ii;
        sTT[wave][row] = tt;
    }
    __syncthreads();

    float loss = 0.f;
    if (tid < 16) {
        float fit = 0.f, fii = 0.f, ftt = 0.f;
#pragma unroll
        for (int w = 0; w < NWAVES; ++w) {     // fixed summation order
            fit += sIT[w][tid];
            fii += sII[w][tid];
            ftt += sTT[w][tid];
        }
        const float eps = 1e-12f;
        float t     = fminf(expf(t_prime[0]), 100.0f);
        float denom = fmaxf(sqrtf(fii), eps) * fmaxf(sqrtf(ftt), eps);
        float logit = (fit / denom) * t + bias[0];
        // -logsigmoid(x) = softplus(-x) = max(-x,0) + log1p(exp(-|x|))
        loss = fmaxf(-logit, 0.f) + log1pf(expf(-fabsf(logit)));
    }

    if (wave == 0) {                           // lanes 16-31 hold 0
#pragma unroll
        for (int off = 16; off > 0; off >>= 1)
            loss += __shfl_xor(loss, off, 32);
        if (lane == 0)
            partial[blockIdx.x] = loss;
    }
}

// Single-block, fixed-shape LDS tree reduction: deterministic (no float atomics).
__global__ __launch_bounds__(512)
void siglip_reduce(const float* __restrict__ partial,
                   float* __restrict__ out,
                   int n, float invB)
{
    __shared__ float s[512];
    const int tid = threadIdx.x;
    s[tid] = (tid < n) ? partial[tid] : 0.f;
    __syncthreads();
#pragma unroll
    for (int off = 256; off > 0; off >>= 1) {
        if (tid < off) s[tid] += s[tid + off];
        __syncthreads();
    }
    if (tid == 0)
        out[0] = s[0] * invB;
}

extern "C" void kernel_launch(void* const* d_in, const int* in_sizes, int n_in,
                              void* d_out, int out_size, void* d_ws, size_t ws_size,
                              hipStream_t stream)
{
    const float* img = (const float*)d_in[0];
    const float* txt = (const float*)d_in[1];
    const float* tpr = (const float*)d_in[2];  // t_prime scalar
    const float* bia = (const float*)d_in[3];  // bias scalar

    const int B    = in_sizes[0] / SIG_D;      // 8192
    const int nblk = B / 16;                   // 512 tiles

    float* partial = (float*)d_ws;             // nblk floats of scratch

    siglip_diag_wmma<<<nblk, 32 * NWAVES, 0, stream>>>(img, txt, tpr, bia, partial);
    siglip_reduce<<<1, 512, 0, stream>>>(partial, (float*)d_out, nblk, 1.0f / (float)B);
}